// GATLayer_62577673503055
// MI455X (gfx1250) — compile-verified
//
#include <hip/hip_runtime.h>
#include <hip/hip_bf16.h>

typedef __attribute__((ext_vector_type(2))) float v2f;
typedef __attribute__((ext_vector_type(4))) float v4f;
typedef __attribute__((ext_vector_type(8))) float v8f;
typedef __attribute__((ext_vector_type(4))) int   v4i;

#define B_    4
#define N_    2048
#define DIN_  128
#define DOUT_ 32
#define HH_   8
#define NEGV  (-9000000000000000.0f)

// ---------------------------------------------------------------------------
// Phase 1: Wh[b,h,n,d] = sum_i H[b,n,i] * W[h,i,d]   (one wave per (b,h,n))
//          ei[b,h,n]   = sum_d Wh * a1[h,d]
//          ej[b,h,n]   = sum_d Wh * a2[h,d]
// ---------------------------------------------------------------------------
__global__ void gat_phase1(const float* __restrict__ H,
                           const float* __restrict__ W,
                           const float* __restrict__ a,
                           float* __restrict__ Wh,
                           float* __restrict__ ei,
                           float* __restrict__ ej) {
    const int idx  = blockIdx.x;           // (b*HH + h)*N + n
    const int n    = idx & (N_ - 1);
    const int h    = (idx >> 11) & (HH_ - 1);
    const int b    = idx >> 14;
    const int lane = threadIdx.x;          // d = 0..31 (wave32)

    const float* Hrow = H + ((long)(b * N_ + n)) * DIN_;   // lane-uniform: SMEM
    const float* Wcol = W + (long)h * DIN_ * DOUT_ + lane;

    float acc = 0.0f;
#pragma unroll 8
    for (int i = 0; i < DIN_; ++i)
        acc = fmaf(Hrow[i], Wcol[i * DOUT_], acc);

    Wh[(((long)(b * HH_ + h)) * N_ + n) * DOUT_ + lane] = acc;

    // a is (HH, 2*DOUT, 1): a1 = [h][0..31], a2 = [h][32..63]
    float p = acc * a[h * (2 * DOUT_) + lane];
    float q = acc * a[h * (2 * DOUT_) + DOUT_ + lane];
#pragma unroll
    for (int off = 16; off >= 1; off >>= 1) {
        p += __shfl_xor(p, off, 32);
        q += __shfl_xor(q, off, 32);
    }
    if (lane == 0) {
        ei[(b * HH_ + h) * N_ + n] = p;
        ej[(b * HH_ + h) * N_ + n] = q;
    }
}

// ---------------------------------------------------------------------------
// Phase 2: fused masked-softmax + streaming alpha store + (alpha x Wh) WMMA.
// One wave (32 threads) per (b, h, 16-row n-tile).  4096 waves total.
// ---------------------------------------------------------------------------
__global__ void __launch_bounds__(32)
gat_phase2(const int* __restrict__ adj,
           const float* __restrict__ Wh,
           const float* __restrict__ ej,
           float* __restrict__ hprime,    // [B][N][HH*DOUT]
           float* __restrict__ alpha) {   // [B][HH][N][N]
    __shared__ float s_ej[N_];            // 8 KB
    __shared__ float s_alpha[16 * 132];   // stride 132: 16B rows + A-load clean
    __shared__ float s_wh[128 * 33];      // stride 33: conflict-free B loads
    __shared__ float s_rowmax[16];
    __shared__ float s_rowsinv[16];

    const int blk  = blockIdx.x;
    const int n0   = (blk & 127) * 16;
    const int h    = (blk >> 7) & (HH_ - 1);
    const int b    = blk >> 10;
    const int lane = threadIdx.x;

    const long bh      = (long)(b * HH_ + h);
    const long whBase  = bh * N_ * DOUT_;
    const long adjBase = (long)b * N_ * N_;
    const long aOutB   = (bh * N_ + n0) * N_;

    // stage ej (B128 coalesced)
    for (int i = lane; i < N_ / 4; i += 32)
        *(v4f*)(s_ej + i * 4) = *(const v4f*)(ej + bh * N_ + i * 4);
    __syncthreads();

    // ---- softmax stats per row: max over masked ej, then sum of exp -------
    for (int r = 0; r < 16; ++r) {
        const v4i* arow4 = (const v4i*)(adj + adjBase + (long)(n0 + r) * N_);
        float mx = NEGV;
        for (int i = lane; i < N_ / 4; i += 32) {
            const v4i aa = arow4[i];
            const v4f ev = *(const v4f*)(s_ej + i * 4);
            if (aa.x) mx = fmaxf(mx, ev.x);
            if (aa.y) mx = fmaxf(mx, ev.y);
            if (aa.z) mx = fmaxf(mx, ev.z);
            if (aa.w) mx = fmaxf(mx, ev.w);
        }
#pragma unroll
        for (int off = 16; off >= 1; off >>= 1)
            mx = fmaxf(mx, __shfl_xor(mx, off, 32));
        float s = 0.0f;
        for (int i = lane; i < N_ / 4; i += 32) {
            const v4i aa = arow4[i];
            const v4f ev = *(const v4f*)(s_ej + i * 4);
            if (aa.x) s += __expf(ev.x - mx);
            if (aa.y) s += __expf(ev.y - mx);
            if (aa.z) s += __expf(ev.z - mx);
            if (aa.w) s += __expf(ev.w - mx);
        }
#pragma unroll
        for (int off = 16; off >= 1; off >>= 1)
            s += __shfl_xor(s, off, 32);
        if (lane == 0) {
            s_rowmax[r]  = mx;
            s_rowsinv[r] = (s > 0.0f) ? (1.0f / s) : 0.0f;
        }
    }
    __syncthreads();

    // ---- accumulators: D = 16x32 f32, two 16x16 WMMA tiles ----------------
    v8f c0 = {};  // cols 0..15
    v8f c1 = {};  // cols 16..31

    const int rA = lane & 15;               // A-matrix row for this lane
    const int kk = (lane >> 4) << 1;        // K sub-offset: 0 / 2
    const int nB = lane & 15;               // B-matrix column
    const int c4 = lane * 4;                // 4 alpha cols per lane

    for (int m0 = 0; m0 < N_; m0 += 128) {
        // stage Wh supertile [m0..m0+127][0..31] (B128 global reads)
        for (int i = lane; i < (128 * DOUT_) / 4; i += 32) {
            const v4f wv = *(const v4f*)(Wh + whBase + (long)m0 * DOUT_ + i * 4);
            const int row = (i * 4) >> 5;
            const int col = (i * 4) & 31;
            s_wh[row * 33 + col + 0] = wv.x;
            s_wh[row * 33 + col + 1] = wv.y;
            s_wh[row * 33 + col + 2] = wv.z;
            s_wh[row * 33 + col + 3] = wv.w;
        }

        // alpha tile: B128 adj load, B128 non-temporal HBM store, LDS copy
        for (int r = 0; r < 16; ++r) {
            const v4i aa =
                *(const v4i*)(adj + adjBase + (long)(n0 + r) * N_ + m0 + c4);
            const v4f ev  = *(const v4f*)(s_ej + m0 + c4);
            const float mxr = s_rowmax[r];
            const float sir = s_rowsinv[r];
            v4f av;
            av.x = aa.x ? __expf(ev.x - mxr) * sir : 0.0f;
            av.y = aa.y ? __expf(ev.y - mxr) * sir : 0.0f;
            av.z = aa.z ? __expf(ev.z - mxr) * sir : 0.0f;
            av.w = aa.w ? __expf(ev.w - mxr) * sir : 0.0f;
            // streamed once, re-read only from LDS -> keep out of L2 (TH=NT)
            __builtin_nontemporal_store(
                av, (v4f*)(alpha + aOutB + (long)r * N_ + m0 + c4));
            *(v4f*)(&s_alpha[r * 132 + c4]) = av;   // ds_store_b128
        }
        __syncthreads();

        // K-loop: D += A(16x4) x B(4x16), f32 WMMA
        for (int k = 0; k < 128; k += 4) {
            v2f A;
            A.x = s_alpha[rA * 132 + k + kk];          // VGPR0: K = kk
            A.y = s_alpha[rA * 132 + k + kk + 1];      // VGPR1: K = kk+1

            const int rowB = (k + kk) * 33;
            v2f B0, B1;
            B0.x = s_wh[rowB + nB];                    // K=kk,   N=0..15
            B0.y = s_wh[rowB + 33 + nB];               // K=kk+1
            B1.x = s_wh[rowB + 16 + nB];               // K=kk,   N=16..31
            B1.y = s_wh[rowB + 33 + 16 + nB];          // K=kk+1

            c0 = __builtin_amdgcn_wmma_f32_16x16x4_f32(
                     false, A, false, B0, (short)0, c0, false, false);
            c1 = __builtin_amdgcn_wmma_f32_16x16x4_f32(
                     false, A, false, B1, (short)0, c1, false, false);
        }
        __syncthreads();
    }

    // ---- epilogue: relu + store h' at [b][n][h*32 + d] --------------------
    // D layout: VGPR j, lanes 0-15 -> (M=j, N=lane); lanes 16-31 -> (M=j+8)
    const int d    = lane & 15;
    const int roff = (lane >> 4) << 3;
#pragma unroll
    for (int j = 0; j < 8; ++j) {
        const int row = n0 + j + roff;
        const long o  = ((long)(b * N_ + row)) * (HH_ * DOUT_) + h * DOUT_;
        hprime[o + d]      = fmaxf(c0[j], 0.0f);
        hprime[o + 16 + d] = fmaxf(c1[j], 0.0f);
    }
}

// ---------------------------------------------------------------------------
extern "C" void kernel_launch(void* const* d_in, const int* in_sizes, int n_in,
                              void* d_out, int out_size, void* d_ws, size_t ws_size,
                              hipStream_t stream) {
    const float* H   = (const float*)d_in[0];   // (4,2048,128)
    const int*   adj = (const int*)d_in[1];     // (4,2048,2048)
    const float* W   = (const float*)d_in[2];   // (8,128,32)
    const float* a   = (const float*)d_in[3];   // (8,64,1)

    float* out    = (float*)d_out;
    float* hprime = out;                                    // 4*2048*256
    float* alpha  = out + (long)B_ * N_ * HH_ * DOUT_;      // 4*8*2048*2048

    float* Wh = (float*)d_ws;                               // 2,097,152 f32
    float* ei = Wh + (long)B_ * HH_ * N_ * DOUT_;           // 65,536 f32
    float* ej = ei + B_ * HH_ * N_;                         // 65,536 f32

    gat_phase1<<<B_ * HH_ * N_, 32, 0, stream>>>(H, W, a, Wh, ei, ej);
    gat_phase2<<<B_ * HH_ * (N_ / 16), 32, 0, stream>>>(adj, Wh, ej,
                                                        hprime, alpha);
}